// LinkPredictor_40785009443380
// MI455X (gfx1250) — compile-verified
//
#include <hip/hip_runtime.h>

typedef float v2f __attribute__((ext_vector_type(2)));
typedef float v8f __attribute__((ext_vector_type(8)));

namespace {
constexpr int NG    = 128;   // graphs == macro tiles per side
constexpr int GN    = 128;   // nodes per graph == macro tile size
constexpr int DIM   = 512;   // embedding dim
constexpr int KC    = 32;    // K chunk staged in LDS
constexpr int LDSS  = 36;    // padded LDS row stride in floats (16B aligned, bank-spread)
constexpr int PAIRS = (GN * (GN - 1)) / 2;  // 8128 outputs per graph
}

__device__ __forceinline__ unsigned f32_to_ordered(float f) {
  unsigned u = __float_as_uint(f);
  return (u & 0x80000000u) ? ~u : (u | 0x80000000u);
}
__device__ __forceinline__ float ordered_to_f32(unsigned u) {
  unsigned b = (u & 0x80000000u) ? (u & 0x7FFFFFFFu) : ~u;
  return __uint_as_float(b);
}

__global__ void init_minmax_kernel(unsigned* __restrict__ mm) {
  mm[0] = 0xFFFFFFFFu;  // ordered-min identity (largest ordered value)
  mm[1] = 0x00000000u;  // ordered-max identity (smallest ordered value)
}

// Pass 1: min/max over sims = E @ E^T, tile-symmetric (only j >= i), nothing stored
// except two atomics per workgroup.
__global__ __launch_bounds__(256) void
sims_minmax_kernel(const float* __restrict__ E, unsigned* __restrict__ mm) {
  const int jt = blockIdx.x;  // column macro tile
  const int it = blockIdx.y;  // row macro tile
  if (jt < it) return;        // symmetry: sims[r,c] == sims[c,r]

  __shared__ float ldsA[GN * LDSS];
  __shared__ float ldsB[GN * LDSS];
  __shared__ float wmn[8], wmx[8];

  const int tid  = threadIdx.x;
  const int wave = tid >> 5;
  const int lane = tid & 31;
  const int m16  = lane & 15;        // row/col within 16 for fragments
  const int kk   = (lane >> 4) << 1; // k sub-offset 0 or 2
  const long rbase = (long)it * GN;
  const long cbase = (long)jt * GN;

  v8f zero = {};
  v8f acc[8];
#pragma unroll
  for (int c = 0; c < 8; ++c) acc[c] = zero;

  for (int kc = 0; kc < DIM; kc += KC) {
    // Stage 128x32 chunks of A-rows and B-cols (both are rows of E).
#pragma unroll
    for (int ld = 0; ld < 4; ++ld) {
      int e   = ld * 256 + tid;   // 0..1023 float4 slots
      int row = e >> 3;           // 0..127
      int k4  = (e & 7) << 2;     // 0,4,...,28
      float4 va = *(const float4*)(E + (rbase + row) * DIM + kc + k4);
      *(float4*)(ldsA + row * LDSS + k4) = va;
      float4 vb = *(const float4*)(E + (cbase + row) * DIM + kc + k4);
      *(float4*)(ldsB + row * LDSS + k4) = vb;
    }
    __syncthreads();
#pragma unroll
    for (int ks = 0; ks < KC; ks += 4) {
      v2f a = *(const v2f*)(ldsA + (wave * 16 + m16) * LDSS + ks + kk);
#pragma unroll
      for (int c = 0; c < 8; ++c) {
        v2f b = *(const v2f*)(ldsB + (c * 16 + m16) * LDSS + ks + kk);
        acc[c] = __builtin_amdgcn_wmma_f32_16x16x4_f32(
            false, a, false, b, (short)0, acc[c], false, false);
      }
    }
    __syncthreads();
  }

  // Reduce min/max: lane-local -> wave shuffle -> LDS -> block -> global atomics.
  float mn = 3.402823466e38f, mx = -3.402823466e38f;
#pragma unroll
  for (int c = 0; c < 8; ++c)
#pragma unroll
    for (int v = 0; v < 8; ++v) {
      float x = acc[c][v];
      mn = fminf(mn, x);
      mx = fmaxf(mx, x);
    }
#pragma unroll
  for (int off = 16; off > 0; off >>= 1) {
    mn = fminf(mn, __shfl_xor(mn, off, 32));
    mx = fmaxf(mx, __shfl_xor(mx, off, 32));
  }
  if (lane == 0) { wmn[wave] = mn; wmx[wave] = mx; }
  __syncthreads();
  if (tid == 0) {
#pragma unroll
    for (int w = 1; w < 8; ++w) {
      mn = fminf(mn, wmn[w]);
      mx = fmaxf(mx, wmx[w]);
    }
    atomicMin(&mm[0], f32_to_ordered(mn));
    atomicMax(&mm[1], f32_to_ordered(mx));
  }
}

// Pass 2: recompute the 128 diagonal blocks (one workgroup per graph), normalize,
// and scatter the strict-upper-triangle entries to the triu_indices flat layout.
__global__ __launch_bounds__(256) void
diag_gather_kernel(const float* __restrict__ E, const unsigned* __restrict__ mm,
                   float* __restrict__ out) {
  const int g = blockIdx.x;

  __shared__ float ldsT[GN * LDSS];

  const int tid  = threadIdx.x;
  const int wave = tid >> 5;
  const int lane = tid & 31;
  const int m16  = lane & 15;
  const int kk   = (lane >> 4) << 1;
  const long base = (long)g * GN;

  v8f zero = {};
  v8f acc[8];
#pragma unroll
  for (int c = 0; c < 8; ++c) acc[c] = zero;

  for (int kc = 0; kc < DIM; kc += KC) {
#pragma unroll
    for (int ld = 0; ld < 4; ++ld) {
      int e   = ld * 256 + tid;
      int row = e >> 3;
      int k4  = (e & 7) << 2;
      float4 v = *(const float4*)(E + (base + row) * DIM + kc + k4);
      *(float4*)(ldsT + row * LDSS + k4) = v;
    }
    __syncthreads();
#pragma unroll
    for (int ks = 0; ks < KC; ks += 4) {
      v2f a = *(const v2f*)(ldsT + (wave * 16 + m16) * LDSS + ks + kk);
#pragma unroll
      for (int c = 0; c < 8; ++c) {
        v2f b = *(const v2f*)(ldsT + (c * 16 + m16) * LDSS + ks + kk);
        acc[c] = __builtin_amdgcn_wmma_f32_16x16x4_f32(
            false, a, false, b, (short)0, acc[c], false, false);
      }
    }
    __syncthreads();
  }

  const float mnv   = ordered_to_f32(mm[0]);
  const float mxv   = ordered_to_f32(mm[1]);
  const float scale = 1.0f / (mxv - mnv + 1e-7f);
  float* og = out + (long)g * PAIRS;

  const int rhalf = (lane >> 4) << 3;  // 0 or 8
#pragma unroll
  for (int c = 0; c < 8; ++c) {
    const int col = c * 16 + m16;
#pragma unroll
    for (int v = 0; v < 8; ++v) {
      const int row = wave * 16 + rhalf + v;
      if (col > row) {
        // flat index within np.triu_indices(128, k=1) row-major ordering
        const int p = row * 127 - (row * (row - 1)) / 2 + (col - row - 1);
        og[p] = (acc[c][v] - mnv) * scale;
      }
    }
  }
}

extern "C" void kernel_launch(void* const* d_in, const int* in_sizes, int n_in,
                              void* d_out, int out_size, void* d_ws, size_t ws_size,
                              hipStream_t stream) {
  (void)in_sizes; (void)n_in; (void)out_size; (void)ws_size;
  const float* E = (const float*)d_in[0];  // [N_GRAPHS*G, D] f32
  // d_in[1]/d_in[2] (row_idx/col_idx) are reproduced analytically in pass 2.
  unsigned* mm = (unsigned*)d_ws;          // 2 ordered-uint slots: [min, max]
  float* out = (float*)d_out;

  init_minmax_kernel<<<dim3(1), dim3(1), 0, stream>>>(mm);
  sims_minmax_kernel<<<dim3(NG, NG), dim3(256), 0, stream>>>(E, mm);
  diag_gather_kernel<<<dim3(NG), dim3(256), 0, stream>>>(E, mm, out);
}